// RandomPadding_37563783971541
// MI455X (gfx1250) — compile-verified
//
#include <hip/hip_runtime.h>

// RandomPadding: out[b,c,y,x] = x[b,c, y+dy(b), x+dx(b)], zero fill OOB,
// dy = start_h - pad_top, dx = start_w - pad_left.
// Fixed shapes: b=256, c=3, h=w=224, float32.
//
// Memory-bound: ~308 MB traffic -> ~13.2 us floor at 23.3 TB/s. All structure
// lives in the grid so per-thread ALU is minimal:
//   gridDim.y = b*3+c, gridDim.x = y, threadIdx.x = x4 (float4 column).
// Row validity (srcY) is wave-uniform -> scalar branch. Interior lanes do one
// 4B-aligned global_load_b128 + one nontemporal global_store_b128.

#define B_   256
#define C_   3
#define H_   224
#define W_   224
#define W4_  (W_ / 4)     // 56 float4 per row
#define BLK_ 64           // 2 waves; lanes >= 56 masked off

typedef float v4f __attribute__((ext_vector_type(4)));

__global__ __launch_bounds__(BLK_) void RandomPadding_kernel(
    const float* __restrict__ x,
    const int*   __restrict__ pad_left,
    const int*   __restrict__ pad_top,
    const int*   __restrict__ start_h,
    const int*   __restrict__ start_w,
    float*       __restrict__ out)
{
    const int x4 = threadIdx.x;          // float4 column within row
    const int y  = blockIdx.x;           // output row       (uniform)
    const int bc = blockIdx.y;           // b*3 + c          (uniform)
    const int b  = bc / C_;              // scalar const-div (uniform)

    // Per-batch shift: wave-uniform -> s_load, L2-resident (256 ints each).
    const int dy = start_h[b] - pad_top[b];
    const int dx = start_w[b] - pad_left[b];

    if (x4 >= W4_) return;

    const long planeW = (long)bc * (H_ * W_);      // element base of this plane
    v4f v = (v4f)0.0f;

    const int srcY = y + dy;                       // uniform -> scalar branch
    if ((unsigned)srcY < (unsigned)H_) {
        const float* row = x + planeW + (long)srcY * W_;
        const int x0 = x4 * 4 + dx;
        if (x0 >= 0 && x0 + 3 < W_) {
            // Interior: 4B-aligned 16B load -> global_load_b128.
            __builtin_memcpy(&v, row + x0, sizeof(v4f));
        } else {
            // Edge lanes only (<=2 lanes per wave when dx != 0).
#pragma unroll
            for (int i = 0; i < 4; ++i) {
                const int xi = x0 + i;
                if ((unsigned)xi < (unsigned)W_) v[i] = row[xi];
            }
        }
    }

    // Streaming (single-use) output: non-temporal b128 store.
    v4f* orow = (v4f*)(out + planeW + (long)y * W_);
    __builtin_nontemporal_store(v, orow + x4);
}

extern "C" void kernel_launch(void* const* d_in, const int* in_sizes, int n_in,
                              void* d_out, int out_size, void* d_ws, size_t ws_size,
                              hipStream_t stream) {
    // setup_inputs() order: x, pad_left, pad_right, pad_top, pad_bottom, start_h, start_w
    const float* x        = (const float*)d_in[0];
    const int*   pad_left = (const int*)  d_in[1];
    // d_in[2] = pad_right  (not used by the reference math)
    const int*   pad_top  = (const int*)  d_in[3];
    // d_in[4] = pad_bottom (not used by the reference math)
    const int*   start_h  = (const int*)  d_in[5];
    const int*   start_w  = (const int*)  d_in[6];
    float*       out      = (float*)d_out;

    dim3 grid(H_, B_ * C_, 1);   // (y, b*3+c)
    dim3 block(BLK_, 1, 1);
    RandomPadding_kernel<<<grid, block, 0, stream>>>(x, pad_left, pad_top,
                                                     start_h, start_w, out);
}